// ComplExScore_1872605741812
// MI455X (gfx1250) — compile-verified
//
#include <hip/hip_runtime.h>

typedef __attribute__((ext_vector_type(2))) float v2f;
typedef __attribute__((ext_vector_type(4))) float v4f;
typedef __attribute__((ext_vector_type(8))) float v8f;

#define DIM  256
#define HALF 128

// 16 edges per wave32. Lane l serves edge (l & 15); lanes l and l+16 split the
// 128-wide half-dim into two contiguous 64-element chunks. Per-lane fp32
// partials are reduced across lanes + K-slots with one V_WMMA_F32_16X16X4_F32
// (A = ones -> D[m][n] = sum_k B[k][n] = full per-edge sum, replicated).
__global__ __launch_bounds__(256) void complex_score_wmma_kernel(
    const float* __restrict__ node_emb,
    const float* __restrict__ rel_emb,
    const int* __restrict__ src,
    const int* __restrict__ dst,
    const int* __restrict__ rel_id,
    float* __restrict__ out,
    int n_edges)
{
    const int lane   = threadIdx.x & 31;
    const int waveIb = threadIdx.x >> 5;
    const int waveId = blockIdx.x * (blockDim.x >> 5) + waveIb;

    const int edge_base = waveId * 16;
    const int e_local   = lane & 15;
    int e = edge_base + e_local;
    const bool valid = (e < n_edges);
    // Clamp instead of branching: keeps EXEC all-1s through the WMMA.
    if (!valid) e = n_edges - 1;

    // Edge-index streams are read-once: keep them out of L2 (node table lives there).
    const int si = __builtin_nontemporal_load(src + e);
    const int di = __builtin_nontemporal_load(dst + e);
    const int ri = __builtin_nontemporal_load(rel_id + e);

    // Offsets fit in int32: 100000*256 = 25.6M.
    const float* __restrict__ h = node_emb + si * DIM;
    const float* __restrict__ t = node_emb + di * DIM;
    const float* __restrict__ r = rel_emb  + ri * DIM;

    // Lane covers elements [chunk, chunk+64) of the 128-wide half.
    const int chunk = (lane >> 4) * 64;

    // Two accumulators to shorten FMA dependency chains.
    v4f acc0 = {0.f, 0.f, 0.f, 0.f};
    v4f acc1 = {0.f, 0.f, 0.f, 0.f};
    // Unroll by 8: 48 b128 loads in flight per batch — near the 63-entry
    // LOADcnt limit, at roughly half the register pressure of a full unroll
    // (more resident waves to hide L2 gather latency).
#pragma unroll 8
    for (int j = 0; j < 16; ++j) {
        const int k = chunk + j * 4;
        v4f hr = *(const v4f*)(h + k);
        v4f hi = *(const v4f*)(h + HALF + k);
        v4f tr = *(const v4f*)(t + k);
        v4f ti = *(const v4f*)(t + HALF + k);
        v4f rr = *(const v4f*)(r + k);
        v4f ri = *(const v4f*)(r + HALF + k);
        // rh*rt*rr + ih*it*rr + rh*it*ir - ih*rt*ir
        //   = rr*(rh*rt + ih*it) + ir*(rh*it - ih*rt)
        v4f term = rr * (hr * tr + hi * ti) + ri * (hr * ti - hi * tr);
        if (j & 1) acc1 += term; else acc0 += term;
    }
    v4f acc = acc0 + acc1;

    // Cross-lane reduction via matrix core. Edge n's partials live in lanes n
    // and n+16, spread over both K-slots of B; A = ones sums all of K.
    v2f a = {1.0f, 1.0f};
    v2f b = {acc.x + acc.y, acc.z + acc.w};
    v8f c = {0.f, 0.f, 0.f, 0.f, 0.f, 0.f, 0.f, 0.f};
    c = __builtin_amdgcn_wmma_f32_16x16x4_f32(
        /*neg_a=*/false, a, /*neg_b=*/false, b,
        /*c_mod=*/(short)0, c, /*reuse_a=*/false, /*reuse_b=*/false);

    // c[0] at lane l holds the sum for edge (l & 15); lanes 0..15 store.
    // Output is write-once: non-temporal so it doesn't evict the node table.
    if (lane < 16 && valid) {
        __builtin_nontemporal_store(c[0], out + e);
    }
}

extern "C" void kernel_launch(void* const* d_in, const int* in_sizes, int n_in,
                              void* d_out, int out_size, void* d_ws, size_t ws_size,
                              hipStream_t stream) {
    const float* node_emb = (const float*)d_in[0];
    const float* rel_emb  = (const float*)d_in[1];
    const int*   src      = (const int*)d_in[2];
    const int*   dst      = (const int*)d_in[3];
    const int*   rel_id   = (const int*)d_in[4];
    float*       out      = (float*)d_out;

    const int n_edges = in_sizes[2];
    if (n_edges <= 0) return;

    const int threads = 256;                         // 8 waves
    const int edges_per_block = (threads / 32) * 16; // 128 edges/block
    const int blocks = (n_edges + edges_per_block - 1) / edges_per_block;

    complex_score_wmma_kernel<<<blocks, threads, 0, stream>>>(
        node_emb, rel_emb, src, dst, rel_id, out, n_edges);
}